// ModelGAT_76192719831693
// MI455X (gfx1250) — compile-verified
//
#include <hip/hip_runtime.h>
#include <cstddef>

typedef __attribute__((ext_vector_type(2))) float v2f;
typedef __attribute__((ext_vector_type(8))) float v8f;

#define NEG_SLOPE 0.2f

__device__ __forceinline__ void atomic_max_float(float* addr, float val) {
    // signed-int max for non-negative floats, unsigned min for negative floats
    if (val >= 0.0f) atomicMax((int*)addr, __float_as_int(val));
    else             atomicMin((unsigned int*)addr, __float_as_uint(val));
}

// ---------------- elementwise / setup kernels ----------------
__global__ void k_fill(float* p, float v, int n) {
    int i = blockIdx.x * blockDim.x + threadIdx.x;
    if (i < n) p[i] = v;
}

__global__ void k_copy(float* d, const float* s, int n) {
    int i = blockIdx.x * blockDim.x + threadIdx.x;
    if (i < n) d[i] = s[i];
}

// X1[n, 0:64] = hn[n,:], rest 0 (aggr cols filled by atomics later)
__global__ void k_init_x1(const float* __restrict__ hn, float* __restrict__ X1, int N) {
    int i = blockIdx.x * blockDim.x + threadIdx.x;
    if (i >= N * 132) return;
    int n = i / 132, f = i - n * 132;
    X1[i] = (f < 64) ? hn[n * 64 + f] : 0.0f;
}

// agg[dst*agg_ld + f] += feat[src*64 + f]; optionally he_agg[dst] += he[e]
__global__ void k_edge_aggr(const int* __restrict__ src, const int* __restrict__ dst,
                            const float* __restrict__ feat, float* __restrict__ agg, int agg_ld,
                            const float* __restrict__ he, float* __restrict__ he_agg, int total) {
    int i = blockIdx.x * blockDim.x + threadIdx.x;
    if (i >= total) return;
    int e = i >> 6, f = i & 63;
    int s = src[e], d = dst[e];
    atomicAdd(&agg[(size_t)d * agg_ld + f], feat[(size_t)s * 64 + f]);
    if (he_agg != nullptr && f == 0) atomicAdd(&he_agg[d], he[e]);
}

__global__ void k_setcol128(float* __restrict__ X1, const float* __restrict__ he_agg, int N) {
    int n = blockIdx.x * blockDim.x + threadIdx.x;
    if (n < N) X1[(size_t)n * 132 + 128] = he_agg[n];
}

// out[n*HD + c] = bias[c]
__global__ void k_init_bias(float* __restrict__ out, const float* __restrict__ bias,
                            int maskHD, int total) {
    int i = blockIdx.x * blockDim.x + threadIdx.x;
    if (i < total) out[i] = bias[i & maskHD];
}

__global__ void k_relu(float* p, int n) {
    int i = blockIdx.x * blockDim.x + threadIdx.x;
    if (i < n) p[i] = fmaxf(p[i], 0.0f);
}

// el/er[n*H+h] = sum_d z[n,h,d] * al/ar[h,d]
__global__ void k_attn(const float* __restrict__ z, const float* __restrict__ al,
                       const float* __restrict__ ar, float* __restrict__ el,
                       float* __restrict__ er, int N, int H) {
    int i = blockIdx.x * blockDim.x + threadIdx.x;
    if (i >= N * H) return;
    int h = i & (H - 1);
    const float* zp  = z  + (size_t)i * 64;
    const float* alp = al + h * 64;
    const float* arp = ar + h * 64;
    float sl = 0.0f, sr = 0.0f;
#pragma unroll 8
    for (int d = 0; d < 64; ++d) { float zv = zp[d]; sl += zv * alp[d]; sr += zv * arp[d]; }
    el[i] = sl;
    er[i] = sr;
}

// e = leakyrelu(el[src]+er[dst]); store; segment-max into m[dst]
__global__ void k_edge_max(const int* __restrict__ src, const int* __restrict__ dst,
                           const float* __restrict__ el, const float* __restrict__ er,
                           float* __restrict__ eb, float* __restrict__ m,
                           int H, int logH, int total) {
    int i = blockIdx.x * blockDim.x + threadIdx.x;
    if (i >= total) return;
    int e = i >> logH, h = i & (H - 1);
    float v = el[src[e] * H + h] + er[dst[e] * H + h];
    v = v > 0.0f ? v : NEG_SLOPE * v;
    eb[i] = v;
    atomic_max_float(&m[dst[e] * H + h], v);
}

// ex = exp(e - m[dst]); store in place; s[dst] += ex
__global__ void k_edge_expsum(const int* __restrict__ dst, float* __restrict__ eb,
                              const float* __restrict__ m, float* __restrict__ s,
                              int H, int logH, int total) {
    int i = blockIdx.x * blockDim.x + threadIdx.x;
    if (i >= total) return;
    int e = i >> logH, h = i & (H - 1);
    int d = dst[e];
    float v = expf(eb[i] - m[d * H + h]);
    eb[i] = v;
    atomicAdd(&s[d * H + h], v);
}

// out[dst, c] += (ex/s) * z[src, c]
__global__ void k_edge_scatter(const int* __restrict__ src, const int* __restrict__ dst,
                               const float* __restrict__ eb, const float* __restrict__ s,
                               const float* __restrict__ z, float* __restrict__ out,
                               int logHD, int maskHD, int H, long long total) {
    long long i = (long long)blockIdx.x * blockDim.x + threadIdx.x;
    if (i >= total) return;
    int e = (int)(i >> logHD);
    int c = (int)i & maskHD;
    int h = c >> 6;
    int d = dst[e], sn = src[e];
    float sv = s[d * H + h];
    float alpha = eb[e * H + h] / (sv > 0.0f ? sv : 1.0f);
    atomicAdd(&out[(size_t)d * (maskHD + 1) + c], alpha * z[(size_t)sn * (maskHD + 1) + c]);
}

// ht = concat(h5, h2agg, he_agg, pad)
__global__ void k_build_ht(float* __restrict__ ht, const float* __restrict__ h5,
                           const float* __restrict__ h2agg, const float* __restrict__ he_agg, int N) {
    int i = blockIdx.x * blockDim.x + threadIdx.x;
    if (i >= N * 132) return;
    int n = i / 132, f = i - n * 132;
    float v;
    if (f < 64)        v = h5[n * 64 + f];
    else if (f < 128)  v = h2agg[n * 64 + (f - 64)];
    else if (f == 128) v = he_agg[n];
    else               v = 0.0f;
    ht[i] = v;
}

__global__ void k_final(const float* __restrict__ r2, const float* __restrict__ Wr3,
                        const float* __restrict__ br3, float* __restrict__ out, int N) {
    int n = blockIdx.x * blockDim.x + threadIdx.x;
    if (n >= N) return;
    float acc = br3[0];
#pragma unroll 8
    for (int d = 0; d < 64; ++d) acc += r2[(size_t)n * 64 + d] * Wr3[d];
    out[n] = acc;
}

// ---------------- WMMA fp32 GEMM: C[N,ncols] = X[N,K] @ W[K,ncols] (+bias, relu) ----
// One block = 16-row band. A-tile staged in LDS (enables safe in-place z=X@W).
// Each wave computes a 16x16 tile with V_WMMA_F32_16X16X4_F32.
__global__ void k_gemm16(const float* __restrict__ X, int ldx,
                         const float* __restrict__ W, int ldw,
                         float* __restrict__ C, int ldc,
                         int K, int pitch,
                         const float* __restrict__ bias, int do_relu) {
    extern __shared__ float As[];           // [16][pitch]
    const int row0 = blockIdx.x * 16;

    // cooperative, coalesced load of the 16 x K A-tile
    for (int i = threadIdx.x; i < 16 * K; i += blockDim.x) {
        int r = i / K, k = i - r * K;
        As[r * pitch + k] = X[(size_t)(row0 + r) * ldx + k];
    }
    __syncthreads();

    const int wave = threadIdx.x >> 5;
    const int lane = threadIdx.x & 31;
    const int col0 = wave * 16;
    const int mrow = lane & 15;             // M for A, N-col for B/C
    const int kh   = (lane >> 4) << 1;      // 0 (lanes 0-15) or 2 (lanes 16-31)

    v8f acc = {0.f, 0.f, 0.f, 0.f, 0.f, 0.f, 0.f, 0.f};
    const float* Wc = W + (col0 + mrow);
    const float* Ar = As + mrow * pitch + kh;

    for (int k0 = 0; k0 < K; k0 += 4) {
        v2f a, b;
        a.x = Ar[k0];
        a.y = Ar[k0 + 1];
        b.x = Wc[(size_t)(k0 + kh) * ldw];
        b.y = Wc[(size_t)(k0 + kh + 1) * ldw];
        acc = __builtin_amdgcn_wmma_f32_16x16x4_f32(
            /*neg_a=*/false, a, /*neg_b=*/false, b,
            /*c_mod=*/(short)0, acc, /*reuse_a=*/false, /*reuse_b=*/false);
    }

    const int rbase = row0 + ((lane >> 4) << 3);   // +8 rows for lanes 16-31
    const int col   = col0 + mrow;
    float bval = bias ? bias[col] : 0.0f;
#pragma unroll
    for (int v = 0; v < 8; ++v) {
        float val = acc[v] + bval;
        if (do_relu) val = fmaxf(val, 0.0f);
        C[(size_t)(rbase + v) * ldc + col] = val;
    }
}

// ---------------- host orchestration ----------------
extern "C" void kernel_launch(void* const* d_in, const int* in_sizes, int n_in,
                              void* d_out, int out_size, void* d_ws, size_t ws_size,
                              hipStream_t stream) {
    const int*   src = (const int*)d_in[0];
    const int*   dst = (const int*)d_in[1];
    const float* hn  = (const float*)d_in[2];
    const float* he  = (const float*)d_in[3];
    const float* W1  = (const float*)d_in[4];
    const float* al1 = (const float*)d_in[5],  *ar1 = (const float*)d_in[6],  *b1 = (const float*)d_in[7];
    const float* W2  = (const float*)d_in[8];
    const float* al2 = (const float*)d_in[9],  *ar2 = (const float*)d_in[10], *b2 = (const float*)d_in[11];
    const float* W3  = (const float*)d_in[12];
    const float* al3 = (const float*)d_in[13], *ar3 = (const float*)d_in[14], *b3 = (const float*)d_in[15];
    const float* W4  = (const float*)d_in[16];
    const float* al4 = (const float*)d_in[17], *ar4 = (const float*)d_in[18], *b4 = (const float*)d_in[19];
    const float* W5  = (const float*)d_in[20];
    const float* al5 = (const float*)d_in[21], *ar5 = (const float*)d_in[22], *b5 = (const float*)d_in[23];
    const float* Wr1 = (const float*)d_in[24], *br1 = (const float*)d_in[25];
    const float* Wr2 = (const float*)d_in[26], *br2 = (const float*)d_in[27];
    const float* Wr3 = (const float*)d_in[28], *br3 = (const float*)d_in[29];

    const int E = in_sizes[0];
    const int N = in_sizes[2] / 64;          // 20000 (divisible by 16)

    float* ws = (float*)d_ws;
    size_t off = 0;
    auto alloc = [&](size_t cnt) { float* p = ws + off; off += cnt; return p; };
    float* X1     = alloc((size_t)N * 132);
    float* A      = alloc((size_t)N * 256);
    float* B      = alloc((size_t)N * 256);
    float* he_agg = alloc((size_t)N);
    float* el     = alloc((size_t)N * 4);
    float* er     = alloc((size_t)N * 4);
    float* mb     = alloc((size_t)N * 4);
    float* sb     = alloc((size_t)N * 4);
    float* eb     = alloc((size_t)E * 4);
    float* W1p    = alloc(132 * 256);
    float* Wr1p   = alloc(132 * 64);

    auto cdiv = [](long long a, int b) { return (unsigned)((a + b - 1) / b); };

    auto run_gemm = [&](const float* X, int ldx, const float* W, int ncols,
                        float* C, int ldc, int K, const float* bias, int relu) {
        int block = (ncols / 16) * 32;
        size_t shmem = (size_t)16 * (K + 1) * sizeof(float);
        k_gemm16<<<N / 16, block, shmem, stream>>>(X, ldx, W, ncols, C, ldc, K, K + 1, bias, relu);
    };

    auto run_conv = [&](const float* X, int ldx, const float* W, int K,
                        const float* al, const float* ar, const float* bias,
                        int H, float* z, float* out, int relu) {
        int HD = H * 64;
        int logH = (H == 4) ? 2 : 0;
        int logHD = (HD == 256) ? 8 : 6;
        run_gemm(X, ldx, W, HD, z, HD, K, nullptr, 0);                       // z = X@W
        k_attn<<<cdiv((long long)N * H, 256), 256, 0, stream>>>(z, al, ar, el, er, N, H);
        k_fill<<<cdiv((long long)N * H, 256), 256, 0, stream>>>(mb, -3.0e38f, N * H);
        k_fill<<<cdiv((long long)N * H, 256), 256, 0, stream>>>(sb, 0.0f, N * H);
        k_init_bias<<<cdiv((long long)N * HD, 256), 256, 0, stream>>>(out, bias, HD - 1, N * HD);
        k_edge_max<<<cdiv((long long)E * H, 256), 256, 0, stream>>>(src, dst, el, er, eb, mb, H, logH, E * H);
        k_edge_expsum<<<cdiv((long long)E * H, 256), 256, 0, stream>>>(dst, eb, mb, sb, H, logH, E * H);
        k_edge_scatter<<<cdiv((long long)E * HD, 256), 256, 0, stream>>>(src, dst, eb, sb, z, out,
                                                                         logHD, HD - 1, H, (long long)E * HD);
        if (relu) k_relu<<<cdiv((long long)N * HD, 256), 256, 0, stream>>>(out, N * HD);
    };

    // ---- stage 0: X1 = [hn | seg_sum(hn[src]) | seg_sum(he) | pad3] ----
    k_init_x1<<<cdiv((long long)N * 132, 256), 256, 0, stream>>>(hn, X1, N);
    k_fill<<<cdiv(N, 256), 256, 0, stream>>>(he_agg, 0.0f, N);
    k_edge_aggr<<<cdiv((long long)E * 64, 256), 256, 0, stream>>>(src, dst, hn, X1 + 64, 132, he, he_agg, E * 64);
    k_setcol128<<<cdiv(N, 256), 256, 0, stream>>>(X1, he_agg, N);

    // ---- pad 129-row weights to 132 ----
    k_fill<<<cdiv(132 * 256, 256), 256, 0, stream>>>(W1p, 0.0f, 132 * 256);
    k_copy<<<cdiv(129 * 256, 256), 256, 0, stream>>>(W1p, W1, 129 * 256);
    k_fill<<<cdiv(132 * 64, 256), 256, 0, stream>>>(Wr1p, 0.0f, 132 * 64);
    k_copy<<<cdiv(129 * 64, 256), 256, 0, stream>>>(Wr1p, Wr1, 129 * 64);

    // ---- GAT stack (z buffers ping-pong; layers 2-4 compute z in place) ----
    run_conv(X1, 132, W1p, 132, al1, ar1, b1, 4, A, B, 1);   // h1 = B
    run_conv(B, 256, W2, 256, al2, ar2, b2, 4, B, A, 1);     // h2 = A (z in-place in B)
    run_conv(A, 256, W3, 256, al3, ar3, b3, 4, A, B, 1);     // h3 = B
    run_conv(B, 256, W4, 256, al4, ar4, b4, 4, B, A, 1);     // h4 = A
    float* z5    = B;
    float* h5    = B + (size_t)N * 64;
    float* h2agg = B + (size_t)2 * N * 64;
    run_conv(A, 256, W5, 256, al5, ar5, b5, 1, z5, h5, 0);   // h5, no relu

    // ---- regression branch ----
    k_fill<<<cdiv((long long)N * 64, 256), 256, 0, stream>>>(h2agg, 0.0f, N * 64);
    k_edge_aggr<<<cdiv((long long)E * 64, 256), 256, 0, stream>>>(src, dst, h5, h2agg, 64, nullptr, nullptr, E * 64);
    float* ht = X1;                                          // reuse (ld 132)
    k_build_ht<<<cdiv((long long)N * 132, 256), 256, 0, stream>>>(ht, h5, h2agg, he_agg, N);
    float* r1 = A;
    float* r2 = A + (size_t)N * 64;
    run_gemm(ht, 132, Wr1p, 64, r1, 64, 132, br1, 1);
    run_gemm(r1, 64, Wr2, 64, r2, 64, 64, br2, 1);
    k_final<<<cdiv(N, 256), 256, 0, stream>>>(r2, Wr3, br3, (float*)d_out, N);
}